// PrototypeLearning_64750926954868
// MI455X (gfx1250) — compile-verified
//
#include <hip/hip_runtime.h>
#include <hip/hip_bf16.h>

// CDNA5 / gfx1250: wave32, WMMA fp32 16x16x4 path, 4-way B-fragment reuse.
typedef float v2f __attribute__((ext_vector_type(2)));
typedef float v8f __attribute__((ext_vector_type(8)));

#define NROWS   131072
#define DIM     512
#define KPROTO  512
#define XS      516   // padded LDS row stride (dwords): lane L banks {4L,4L+1}, L+16 -> {4L+2,4L+3}: conflict-free b64
#define TILES   4     // 16-row A tiles per block -> 64 rows/block
#define ROWS_PB (16 * TILES)

// ---------------------------------------------------------------------------
// Kernel 1: pnorm[k] = 0.5 * ||proto_k||^2 ; also zero the f64 loss accumulator.
// ---------------------------------------------------------------------------
__global__ __launch_bounds__(256)
void vq_pnorm_kernel(const float* __restrict__ protos,
                     float* __restrict__ pnorm,
                     double* __restrict__ acc) {
    const int wave = threadIdx.x >> 5;
    const int lane = threadIdx.x & 31;
    const int row  = blockIdx.x * 8 + wave;

    const float* p = protos + (size_t)row * DIM;
    float s = 0.0f;
    #pragma unroll 4
    for (int i = lane; i < DIM; i += 32) {
        float v = p[i];
        s += v * v;
    }
    #pragma unroll
    for (int off = 16; off > 0; off >>= 1)
        s += __shfl_xor(s, off, 32);
    if (lane == 0) pnorm[row] = 0.5f * s;
    if (blockIdx.x == 0 && threadIdx.x == 0) *acc = 0.0;  // stream-ordered before main kernel
}

// ---------------------------------------------------------------------------
// Kernel 2: main VQ kernel. One block = 64 rows of x (4 WMMA row tiles).
// 8 waves; each wave owns one 16-proto column chunk per pass and reuses every
// B fragment across the 4 A tiles (4 v_wmma per B load).
// ---------------------------------------------------------------------------
__global__ __launch_bounds__(256)
void vq_main_kernel(const float* __restrict__ x,
                    const float* __restrict__ protos,
                    const float* __restrict__ pnorm,
                    float* __restrict__ outq,
                    double* __restrict__ acc) {
    __shared__ float x_lds[TILES * 16 * XS];     // 132,096 B
    __shared__ float bestv_lds[8 * ROWS_PB];     //   2,048 B  [wave][row]
    __shared__ int   besti_lds[8 * ROWS_PB];     //   2,048 B
    __shared__ int   idx_lds[ROWS_PB];
    __shared__ float lds_sum;

    const int t      = threadIdx.x;
    const int wave   = t >> 5;
    const int lane   = t & 31;
    const int laneLo = lane & 15;   // M (A) / N (B) index
    const int laneHi = lane >> 4;   // selects K pair {0,1} vs {2,3}
    const int row0   = blockIdx.x * ROWS_PB;

    if (t == 0) lds_sum = 0.0f;

    // ---- stage x tile (64 x 512 f32) into padded LDS, coalesced float4 loads
    const float4* xg = (const float4*)(x + (size_t)row0 * DIM);
    for (int i = t; i < ROWS_PB * DIM / 4; i += 256) {
        float4 v = xg[i];
        int row = i >> 7;                 // 128 float4 per row
        int c   = (i & 127) << 2;
        float* dst = &x_lds[(row >> 4) * (16 * XS) + (row & 15) * XS + c];
        dst[0] = v.x; dst[1] = v.y; dst[2] = v.z; dst[3] = v.w;
    }
    __syncthreads();

    float bestv = -3.402823466e38f;      // running per-row best (threads 0..63)
    int   besti = 0;

    // A-fragment bases, one per 16-row tile:
    // lane L<16 reads x[M=L][d..d+1], lane L+16 reads x[M=L][d+2..d+3]
    const float* xl0 = &x_lds[0 * (16 * XS) + laneLo * XS + (laneHi << 1)];
    const float* xl1 = &x_lds[1 * (16 * XS) + laneLo * XS + (laneHi << 1)];
    const float* xl2 = &x_lds[2 * (16 * XS) + laneLo * XS + (laneHi << 1)];
    const float* xl3 = &x_lds[3 * (16 * XS) + laneLo * XS + (laneHi << 1)];

    for (int pass = 0; pass < 4; ++pass) {
        const int n0 = pass * 128 + wave * 16;
        const float* bp = protos + (size_t)(n0 + laneLo) * DIM + (laneHi << 1);

        v8f acc0 = {}, acc1 = {}, acc2 = {}, acc3 = {};
        #pragma unroll 4
        for (int d = 0; d < DIM; d += 4) {
            v2f b  = *(const v2f*)(bp  + d);   // one B fragment ...
            v2f a0 = *(const v2f*)(xl0 + d);   // ... reused by 4 row tiles
            v2f a1 = *(const v2f*)(xl1 + d);
            v2f a2 = *(const v2f*)(xl2 + d);
            v2f a3 = *(const v2f*)(xl3 + d);
            acc0 = __builtin_amdgcn_wmma_f32_16x16x4_f32(false, a0, false, b, (short)0, acc0, false, false);
            acc1 = __builtin_amdgcn_wmma_f32_16x16x4_f32(false, a1, false, b, (short)0, acc1, false, false);
            acc2 = __builtin_amdgcn_wmma_f32_16x16x4_f32(false, a2, false, b, (short)0, acc2, false, false);
            acc3 = __builtin_amdgcn_wmma_f32_16x16x4_f32(false, a3, false, b, (short)0, acc3, false, false);
        }

        // score = x.p - 0.5*||p||^2 ; argmax(score) == argmin(distance).
        // In-wave butterfly argmax across the 16 N-lanes of each half-wave,
        // min-index on ties (matches jnp.argmin first-occurrence).
        // NOTE: bitwise | / & on materialized bools -> branchless v_cmp+v_cndmask,
        // no exec-mask manipulation.
        const float pn  = pnorm[n0 + laneLo];
        const int   myi = n0 + laneLo;
        #pragma unroll
        for (int tile = 0; tile < TILES; ++tile) {
            const v8f a = (tile == 0) ? acc0 : (tile == 1) ? acc1 : (tile == 2) ? acc2 : acc3;
            #pragma unroll
            for (int r = 0; r < 8; ++r) {
                float v = a[r] - pn;
                int   i = myi;
                #pragma unroll
                for (int off = 1; off <= 8; off <<= 1) {   // stays within 16-lane half
                    float v2 = __shfl_xor(v, off, 32);
                    int   i2 = __shfl_xor(i, off, 32);
                    int gt = (int)(v2 > v);
                    int eq = (int)(v2 == v);
                    int li = (int)(i2 < i);
                    int take = gt | (eq & li);
                    v = take ? v2 : v;
                    i = take ? i2 : i;
                }
                if (laneLo == 0) {        // lanes 0 (M=r) and 16 (M=r+8)
                    int row = tile * 16 + r + (laneHi << 3);
                    bestv_lds[wave * ROWS_PB + row] = v;
                    besti_lds[wave * ROWS_PB + row] = i;
                }
            }
        }
        __syncthreads();

        // threads 0..63: fold the 8 per-wave candidates (ascending proto index,
        // strict '>' keeps the first/lowest index on ties)
        if (t < ROWS_PB) {
            #pragma unroll
            for (int w2 = 0; w2 < 8; ++w2) {
                float v = bestv_lds[w2 * ROWS_PB + t];
                int   i = besti_lds[w2 * ROWS_PB + t];
                int take = (int)(v > bestv);
                bestv = take ? v : bestv;
                besti = take ? i : besti;
            }
        }
        __syncthreads();   // bestv_lds reused next pass
    }

    if (t < ROWS_PB) idx_lds[t] = besti;
    __syncthreads();

    // ---- gather codebook rows, write quantized output, accumulate loss
    float lsum = 0.0f;
    for (int i = t; i < ROWS_PB * DIM; i += 256) {
        int r = i >> 9;
        int c = i & (DIM - 1);
        float pv = protos[(size_t)idx_lds[r] * DIM + c];
        float xv = x_lds[(r >> 4) * (16 * XS) + (r & 15) * XS + c];
        outq[(size_t)(row0 + r) * DIM + c] = pv;   // quantized_st == quantized numerically
        float dd = pv - xv;
        lsum += dd * dd;
    }
    atomicAdd(&lds_sum, lsum);                      // ds_add_f32
    __syncthreads();
    if (t == 0) atomicAdd(acc, (double)lds_sum);    // global_atomic_add_f64
}

// ---------------------------------------------------------------------------
// Kernel 3: copy prototypes to output tail; write scalar loss.
// ---------------------------------------------------------------------------
__global__ __launch_bounds__(256)
void vq_finalize_kernel(const float* __restrict__ protos,
                        const double* __restrict__ acc,
                        float* __restrict__ out) {
    const size_t NQ = (size_t)NROWS * DIM;
    size_t i = (size_t)blockIdx.x * 256 + threadIdx.x;
    if (i < (size_t)KPROTO * DIM) out[NQ + 1 + i] = protos[i];
    if (i == 0) {
        // loss = q_latent + 0.25*e_latent = 1.25 * mean((quantized - x)^2)
        out[NQ] = (float)(1.25 * (*acc) / (double)NQ);
    }
}

extern "C" void kernel_launch(void* const* d_in, const int* in_sizes, int n_in,
                              void* d_out, int out_size, void* d_ws, size_t ws_size,
                              hipStream_t stream) {
    const float* x      = (const float*)d_in[0];   // [131072, 512] f32
    const float* protos = (const float*)d_in[1];   // [512, 512] f32
    float*  out   = (float*)d_out;                 // quantized | loss | prototypes
    double* acc   = (double*)d_ws;                 // f64 loss accumulator
    float*  pnorm = (float*)((char*)d_ws + 64);    // 512 f32 half-norms

    vq_pnorm_kernel   <<<KPROTO / 8,           256, 0, stream>>>(protos, pnorm, acc);
    vq_main_kernel    <<<NROWS / ROWS_PB,      256, 0, stream>>>(x, protos, pnorm, out, acc);
    vq_finalize_kernel<<<(KPROTO * DIM) / 256, 256, 0, stream>>>(protos, acc, out);
}